// CapsuleNetwork_82463372083898
// MI455X (gfx1250) — compile-verified
//
#include <hip/hip_runtime.h>
#include <hip/hip_bf16.h>
#include <math.h>

typedef float v2f __attribute__((ext_vector_type(2)));
typedef float v4f __attribute__((ext_vector_type(4)));
typedef float v8f __attribute__((ext_vector_type(8)));

#define B_    16
#define CIN   103
#define CCONV 256
#define NOUT  103
#define DOUT  16
#define FDIM  1648      // NOUT*DOUT
#define RECN  8343      // 103*9*9
#define H1_   5562
#define H2_   12514

// ---------------------------------------------------------------------------
// Conv 3x3 VALID + bias + ReLU + spatial mean -> p[b][co]
// grid: B_*CCONV blocks of 64 threads; block = (b, co)
// ---------------------------------------------------------------------------
__global__ void conv_mean_kernel(const float* __restrict__ x,
                                 const float* __restrict__ w,
                                 const float* __restrict__ bias,
                                 float* __restrict__ p)
{
    __shared__ __align__(16) float sx[CIN * 81];
    __shared__ __align__(16) float sw[CIN * 9];
    __shared__ float red[64];
    const int blk = blockIdx.x;
    const int b   = blk >> 8;
    const int co  = blk & 255;
    const int tid = threadIdx.x;

    for (int i = tid; i < CIN * 81; i += 64) sx[i] = x[b * CIN * 81 + i];
    for (int i = tid; i < CIN * 9;  i += 64) sw[i] = w[co * CIN * 9 + i];
    __syncthreads();

    float r = 0.0f;
    if (tid < 49) {
        const int oh = tid / 7, ow = tid % 7;
        float acc = 0.0f;
        for (int ci = 0; ci < CIN; ci++) {
            const float* xr = &sx[ci * 81 + oh * 9 + ow];
            const float* wr = &sw[ci * 9];
            acc += xr[0]  * wr[0] + xr[1]  * wr[1] + xr[2]  * wr[2]
                 + xr[9]  * wr[3] + xr[10] * wr[4] + xr[11] * wr[5]
                 + xr[18] * wr[6] + xr[19] * wr[7] + xr[20] * wr[8];
        }
        r = fmaxf(acc + bias[co], 0.0f);
    }
    red[tid] = r;
    __syncthreads();
    for (int st = 32; st > 0; st >>= 1) {
        if (tid < st) red[tid] += red[tid + st];
        __syncthreads();
    }
    if (tid == 0) p[b * CCONV + co] = red[0] * (1.0f / 49.0f);
}

// ---------------------------------------------------------------------------
// sp[b][c] = squash(p[b,:])[c]   (one block, 256 threads)
// ---------------------------------------------------------------------------
__global__ void squash_p_kernel(const float* __restrict__ p, float* __restrict__ sp)
{
    __shared__ float red[256];
    __shared__ float fac[16];
    const int tid = threadIdx.x;
    const int b = tid >> 4, part = tid & 15;
    float ps = 0.f;
    for (int c = part * 16; c < part * 16 + 16; c++) { float t = p[b * 256 + c]; ps += t * t; }
    red[tid] = ps;
    __syncthreads();
    if (part == 0) {
        float ms = 0.f;
        for (int k = 0; k < 16; k++) ms += red[b * 16 + k];
        fac[b] = sqrtf(ms) / (1.0f + ms);   // mag_sq/(1+mag_sq)/mag
    }
    __syncthreads();
    const float f = fac[b];
    for (int c = part * 16; c < part * 16 + 16; c++) sp[b * 256 + c] = f * p[b * 256 + c];
}

// ---------------------------------------------------------------------------
// Wsum[c][j][o] = sum_i W_caps[c][j][o][i]   (421888 outputs)
// ---------------------------------------------------------------------------
__global__ void wsum_kernel(const float* __restrict__ Wcaps, float* __restrict__ Wsum)
{
    const int gid = blockIdx.x * 256 + threadIdx.x;
    const float* src = Wcaps + (size_t)gid * 32;
    float s = 0.f;
    for (int i = 0; i < 32; i++) s += src[i];
    Wsum[gid] = s;
}

__global__ void binit_kernel(float* __restrict__ b_ij)
{
    b_ij[blockIdx.x * 256 + threadIdx.x] = 1.0f;
}

// ---------------------------------------------------------------------------
// Routing: per output capsule j:
//   cij[c] = softmax_c(b_ij[j][c])
//   s[b][j][o] = sum_c sp[b][c] * (cij[c]*Wsum[c][j][o])   via V_WMMA_F32_16X16X4_F32
// grid: 103 blocks x 256 threads (wave 0 runs the WMMA K-loop)
// ---------------------------------------------------------------------------
__global__ void routing_s_kernel(const float* __restrict__ sp,
                                 const float* __restrict__ Wsum,
                                 const float* __restrict__ b_ij,
                                 float* __restrict__ s_out)
{
    __shared__ float red[256];
    __shared__ __align__(16) float sA[16 * 260];   // sp, row stride 260 (bank spread)
    __shared__ __align__(16) float sB[256 * 16];   // cij[c]*Wsum[c][j][o]
    const int j = blockIdx.x;
    const int tid = threadIdx.x;           // == c

    // softmax over c of b_ij[j][:]
    const float bv = b_ij[j * 256 + tid];
    red[tid] = bv;
    __syncthreads();
    for (int st = 128; st > 0; st >>= 1) {
        if (tid < st) red[tid] = fmaxf(red[tid], red[tid + st]);
        __syncthreads();
    }
    const float mx = red[0];
    __syncthreads();
    const float e = expf(bv - mx);
    red[tid] = e;
    __syncthreads();
    for (int st = 128; st > 0; st >>= 1) {
        if (tid < st) red[tid] += red[tid + st];
        __syncthreads();
    }
    const float cw = e / red[0];

    // stage A (sp) and B (cij*Wsum slice) in LDS
    for (int r = 0; r < 16; r++) sA[r * 260 + tid] = sp[r * 256 + tid];
    const float* wp = Wsum + ((size_t)tid * 103 + j) * 16;
    for (int o = 0; o < 16; o++) sB[tid * 16 + o] = cw * wp[o];
    __syncthreads();

    if (tid < 32) {
        const int lane = tid;
        const int m    = lane & 15;
        const int col  = lane & 15;
        const int koff = (lane >> 4) << 1;      // 0 or 2
        v8f acc = {0.f, 0.f, 0.f, 0.f, 0.f, 0.f, 0.f, 0.f};
        for (int k0 = 0; k0 < 256; k0 += 4) {
            v2f a = *(const v2f*)&sA[m * 260 + k0 + koff];
            v2f bfrag;
            bfrag.x = sB[(k0 + koff) * 16 + col];
            bfrag.y = sB[(k0 + koff + 1) * 16 + col];
            acc = __builtin_amdgcn_wmma_f32_16x16x4_f32(
                false, a, false, bfrag, (short)0, acc, false, false);
        }
        const int rbase = (lane >> 4) << 3;     // 0 or 8
        for (int r = 0; r < 8; r++)
            s_out[(r + rbase) * FDIM + j * 16 + col] = acc[r];
    }
}

// ---------------------------------------------------------------------------
// v[b][j][o] = squash over j:  ms[b][o] = sum_j s^2 ; v = sqrt(ms)/(1+ms) * s
// one block, 256 threads = (b,o)
// ---------------------------------------------------------------------------
__global__ void squash_v_kernel(const float* __restrict__ s, float* __restrict__ v)
{
    const int tid = threadIdx.x;
    const int b = tid >> 4, o = tid & 15;
    const float* sp_ = s + b * FDIM + o;
    float ms = 0.f;
    for (int j = 0; j < NOUT; j++) { float t = sp_[j * 16]; ms += t * t; }
    const float factor = sqrtf(ms) / (1.0f + ms);
    float* vp = v + b * FDIM + o;
    for (int j = 0; j < NOUT; j++) vp[j * 16] = factor * sp_[j * 16];
}

// ---------------------------------------------------------------------------
// b_ij[j][c] += (1/16) * sum_b sp[b][c] * sum_o Wsum[c][j][o]*v[b][j][o]
// grid: 103 blocks x 256 threads (thread == c)
// ---------------------------------------------------------------------------
__global__ void update_b_kernel(const float* __restrict__ sp,
                                const float* __restrict__ Wsum,
                                const float* __restrict__ v,
                                float* __restrict__ b_ij)
{
    __shared__ float vj[256];
    const int j = blockIdx.x;
    const int c = threadIdx.x;
    vj[c] = v[(c >> 4) * FDIM + j * 16 + (c & 15)];
    __syncthreads();
    float wv[16];
    const float* wp = Wsum + ((size_t)c * 103 + j) * 16;
    for (int o = 0; o < 16; o++) wv[o] = wp[o];
    float acc = 0.f;
    for (int b = 0; b < 16; b++) {
        float inner = 0.f;
        for (int o = 0; o < 16; o++) inner += wv[o] * vj[b * 16 + o];
        acc += sp[b * 256 + c] * inner;
    }
    b_ij[j * 256 + c] += acc * (1.0f / 16.0f);
}

// ---------------------------------------------------------------------------
// ba[b][j] = ||v[b][j][:]|| ; f[b][j*16+o] = v * ba
// ---------------------------------------------------------------------------
__global__ void final_ba_f_kernel(const float* __restrict__ v,
                                  float* __restrict__ ba_out,
                                  float* __restrict__ f)
{
    const int gid = blockIdx.x * 256 + threadIdx.x;
    if (gid >= B_ * NOUT) return;
    const int b = gid / NOUT, j = gid % NOUT;
    const float* vp = v + b * FDIM + j * 16;
    float ss = 0.f;
    for (int o = 0; o < 16; o++) ss += vp[o] * vp[o];
    const float n = sqrtf(ss);
    ba_out[b * NOUT + j] = n;
    float* fp = f + b * FDIM + j * 16;
    for (int o = 0; o < 16; o++) fp[o] = vp[o] * n;
}

// ---------------------------------------------------------------------------
// C(16xN) = A(16xK) @ Bw(KxN) + bias, fp32 WMMA 16x16x4.
// HBM-bound (weights ~720MB stream once; AI = 8 FLOP/B at 23.3 TB/s).
//   - Block = 256 threads = 8 waves, covers 64 output columns.
//   - Each wave owns 4 interleaved 16x16 WMMA tiles (tile t -> cols base+4q+t),
//     so each lane pulls a contiguous float4 per B row (global_load_b128,
//     512B per load instruction per wave) and the A fragment is reused x4.
//   - 8-way K-split across the waves of a block (disjoint K-slices of each
//     LDS A chunk) -> 8x more concurrent HBM streams; deterministic in-order LDS
//     reduction (+bias) at the end, no atomics.
// ---------------------------------------------------------------------------
#define ACHUNK  512
#define ASTRIDE 516   // bank-spread padding

__global__ void gemm16_wmma(const float* __restrict__ A,
                            const float* __restrict__ Bw,
                            const float* __restrict__ bias,
                            float* __restrict__ C,
                            int K, int N)
{
    __shared__ __align__(16) float sA[16 * ASTRIDE];   // 33 KB
    __shared__ __align__(16) float part[8 * 16 * 64];  // 32 KB wave partials
    const int tid  = threadIdx.x;
    const int lane = tid & 31;
    const int wave = tid >> 5;                  // 0..7 = K-slice id
    const int q    = lane & 15;
    const int m    = q;
    const int koff = (lane >> 4) << 1;          // 0 or 2
    const int base = blockIdx.x * 64;
    const int col4 = base + 4 * q;              // first of 4 cols this lane loads
    const bool fullN = (base + 64 <= N);

    v8f acc[4];
    for (int t = 0; t < 4; t++) acc[t] = (v8f){0.f,0.f,0.f,0.f,0.f,0.f,0.f,0.f};

    for (int kc = 0; kc < K; kc += ACHUNK) {
        const int kcnt = min(ACHUNK, K - kc);
        // cooperative A chunk load (zero-padded to the chunk)
        for (int idx = tid; idx < 16 * ACHUNK; idx += 256) {
            const int r  = idx >> 9;            // / ACHUNK
            const int kk = idx & (ACHUNK - 1);
            sA[r * ASTRIDE + kk] = (kk < kcnt) ? A[r * K + kc + kk] : 0.0f;
        }
        __syncthreads();

        // this wave's K-slice within the chunk
        const int kpad = (kcnt + 3) & ~3;
        const int kend = min(wave * 64 + 64, kpad);
        for (int k0 = wave * 64; k0 < kend; k0 += 4) {
            v2f a = *(const v2f*)&sA[m * ASTRIDE + k0 + koff];
            const int kg = kc + k0 + koff;
            v4f b0 = {0.f,0.f,0.f,0.f}, b1 = {0.f,0.f,0.f,0.f};
            if (fullN) {
                // memcpy form: wide load without a false 16B-alignment promise
                if (kg     < K) __builtin_memcpy(&b0, Bw + (size_t)kg       * N + col4, 16);
                if (kg + 1 < K) __builtin_memcpy(&b1, Bw + (size_t)(kg + 1) * N + col4, 16);
            } else {
                for (int t = 0; t < 4; t++) {
                    if (col4 + t < N) {
                        if (kg     < K) b0[t] = Bw[(size_t)kg       * N + col4 + t];
                        if (kg + 1 < K) b1[t] = Bw[(size_t)(kg + 1) * N + col4 + t];
                    }
                }
            }
            for (int t = 0; t < 4; t++) {
                v2f bf; bf.x = b0[t]; bf.y = b1[t];
                acc[t] = __builtin_amdgcn_wmma_f32_16x16x4_f32(
                    false, a, false, bf, (short)0, acc[t], false, false);
            }
        }
        __syncthreads();
    }

    // spill wave partials: part[wave][row][colo], colo = 4q+t, row = r + 8*(lane>=16)
    const int rbase = (lane >> 4) << 3;
    for (int t = 0; t < 4; t++)
        for (int r = 0; r < 8; r++)
            part[wave * 1024 + (r + rbase) * 64 + 4 * q + t] = acc[t][r];
    __syncthreads();

    // deterministic reduction over the 8 K-slices, add bias, store
    for (int idx = tid; idx < 1024; idx += 256) {
        float sum = 0.f;
        for (int w2 = 0; w2 < 8; w2++) sum += part[w2 * 1024 + idx];
        const int row  = idx >> 6;
        const int colo = idx & 63;
        const int col  = base + colo;
        if (col < N) C[(size_t)row * N + col] = sum + bias[col];
    }
}

// ---------------------------------------------------------------------------
extern "C" void kernel_launch(void* const* d_in, const int* in_sizes, int n_in,
                              void* d_out, int out_size, void* d_ws, size_t ws_size,
                              hipStream_t stream)
{
    const float* x      = (const float*)d_in[0];
    const float* conv_w = (const float*)d_in[1];
    const float* conv_b = (const float*)d_in[2];
    const float* W_caps = (const float*)d_in[3];
    const float* fc1_w  = (const float*)d_in[4];
    const float* fc1_b  = (const float*)d_in[5];
    const float* fc2_w  = (const float*)d_in[6];
    const float* fc2_b  = (const float*)d_in[7];
    const float* fc3_w  = (const float*)d_in[8];
    const float* fc3_b  = (const float*)d_in[9];
    float* out = (float*)d_out;                 // [0..1647]=ba, rest=recon
    float* ws  = (float*)d_ws;

    float* p    = ws;                  // 16*256      = 4096
    float* sp   = p    + 4096;         // 4096
    float* Wsum = sp   + 4096;         // 256*103*16  = 421888
    float* b_ij = Wsum + 421888;       // 103*256     = 26368  ([j][c])
    float* s    = b_ij + 26368;        // 16*1648     = 26368  ([b][j][o])
    float* v    = s    + 26368;        // 26368
    float* f    = v    + 26368;        // 26368
    float* a1   = f    + 26368;        // 16*5562     = 88992
    float* a2   = a1   + 88992;        // 16*12514    = 200224

    conv_mean_kernel<<<B_ * CCONV, 64, 0, stream>>>(x, conv_w, conv_b, p);
    squash_p_kernel<<<1, 256, 0, stream>>>(p, sp);
    wsum_kernel<<<(CCONV * NOUT * DOUT) / 256, 256, 0, stream>>>(W_caps, Wsum);
    binit_kernel<<<(NOUT * CCONV) / 256, 256, 0, stream>>>(b_ij);

    for (int it = 0; it < 3; it++) {
        routing_s_kernel<<<NOUT, 256, 0, stream>>>(sp, Wsum, b_ij, s);
        squash_v_kernel<<<1, 256, 0, stream>>>(s, v);
        if (it < 2)
            update_b_kernel<<<NOUT, 256, 0, stream>>>(sp, Wsum, v, b_ij);
    }

    final_ba_f_kernel<<<(B_ * NOUT + 255) / 256, 256, 0, stream>>>(v, out, f);

    gemm16_wmma<<<(H1_  + 63) / 64, 256, 0, stream>>>(f,  fc1_w, fc1_b, a1,         FDIM, H1_);
    gemm16_wmma<<<(H2_  + 63) / 64, 256, 0, stream>>>(a1, fc2_w, fc2_b, a2,         H1_,  H2_);
    gemm16_wmma<<<(RECN + 63) / 64, 256, 0, stream>>>(a2, fc3_w, fc3_b, out + 1648, H2_,  RECN);
}